// KANCell_67001489818157
// MI455X (gfx1250) — compile-verified
//
#include <hip/hip_runtime.h>

// relu(sum(relu(x))) over N fp32 elements.
// Stage 1: grid-stride streaming with global_load_b128; per-wave accumulation
//          on the matrix pipe via V_WMMA_F32_16X16X4_F32 with an all-ones A
//          (each WMMA folds 64 fp32 values into the f32 accumulator).
// Stage 2: deterministic final tree-reduce of block partials + outer ReLU.

typedef float v2f __attribute__((ext_vector_type(2)));
typedef float v8f __attribute__((ext_vector_type(8)));

#define NBLOCKS  1024
#define NTHREADS 256

__device__ __forceinline__ float wave32_reduce_add(float v) {
    // 5-step xor-shuffle tree across the wave32 (ds_swizzle / v_permlane path)
    v += __shfl_xor(v, 16, 32);
    v += __shfl_xor(v,  8, 32);
    v += __shfl_xor(v,  4, 32);
    v += __shfl_xor(v,  2, 32);
    v += __shfl_xor(v,  1, 32);
    return v;
}

__global__ void kan_reduce_stage1(const float* __restrict__ x,
                                  float* __restrict__ partials,
                                  long long n4) {
    const int tid = threadIdx.x;
    const long long gid    = (long long)blockIdx.x * blockDim.x + tid;
    const long long stride = (long long)gridDim.x * blockDim.x;
    const float4* __restrict__ x4 = (const float4*)x;

    v8f acc0 = {};   // WMMA accumulator chain 0 (elements .x .y)
    v8f acc1 = {};   // WMMA accumulator chain 1 (elements .z .w)
    v2f onesA;
    onesA.x = 1.0f; onesA.y = 1.0f;   // A = 16x4 all-ones -> D[m][n] = sum_k B[k][n] + C

    // Uniform trip count: every lane of every wave runs exactly `iters`
    // iterations, so EXEC is all-1s at each WMMA (ISA requirement).
    const long long iters = n4 / stride;
    long long i = gid;
    for (long long it = 0; it < iters; ++it, i += stride) {
        float4 v = x4[i];                       // global_load_b128, streaming
        v2f b0, b1;
        b0.x = fmaxf(v.x, 0.0f);                // per-element ReLU (v_max_num_f32)
        b0.y = fmaxf(v.y, 0.0f);
        b1.x = fmaxf(v.z, 0.0f);
        b1.y = fmaxf(v.w, 0.0f);
        // D = ones(16x4) x B(4x16) + C : folds 64 fp32 per instruction on the
        // matrix pipe. Two independent chains hide the WMMA RAW latency.
        acc0 = __builtin_amdgcn_wmma_f32_16x16x4_f32(
                   false, onesA, false, b0, (short)0, acc0, false, false);
        acc1 = __builtin_amdgcn_wmma_f32_16x16x4_f32(
                   false, onesA, false, b1, (short)0, acc1, false, false);
    }

    // Remainder float4s (none for N = 2^25, kept for generality; divergence is
    // fine here because no WMMA executes under it).
    float rsum = 0.0f;
    if (i < n4) {
        float4 v = x4[i];
        rsum = fmaxf(v.x, 0.0f) + fmaxf(v.y, 0.0f)
             + fmaxf(v.z, 0.0f) + fmaxf(v.w, 0.0f);
    }

    // acc[0] per lane = one 16-wide column sum; across the 32 lanes each
    // column sum appears exactly twice (M=0 and M=8 rows of D), so the wave
    // total is 0.5 * reduce32(acc0[0] + acc1[0]), plus the scalar remainder.
    float lane_v = 0.5f * (acc0[0] + acc1[0]) + rsum;
    float wsum = wave32_reduce_add(lane_v);

    __shared__ float smem[NTHREADS / 32];
    const int wave = tid >> 5;
    const int lane = tid & 31;
    if (lane == 0) smem[wave] = wsum;
    __syncthreads();
    if (tid == 0) {
        float t = 0.0f;
        #pragma unroll
        for (int w = 0; w < NTHREADS / 32; ++w) t += smem[w];
        partials[blockIdx.x] = t;   // one deterministic partial per block
    }
}

__global__ void kan_reduce_stage2(const float* __restrict__ partials, int nparts,
                                  const float* __restrict__ x,
                                  long long n, long long n4,
                                  float* __restrict__ out) {
    const int tid = threadIdx.x;
    float s = 0.0f;
    for (int i = tid; i < nparts; i += NTHREADS) s += partials[i];
    // Scalar tail elements (n % 4), none for the reference N.
    for (long long i = n4 * 4 + tid; i < n; i += NTHREADS) s += fmaxf(x[i], 0.0f);

    float wsum = wave32_reduce_add(s);
    __shared__ float smem[NTHREADS / 32];
    const int wave = tid >> 5;
    const int lane = tid & 31;
    if (lane == 0) smem[wave] = wsum;
    __syncthreads();
    if (tid == 0) {
        float t = 0.0f;
        #pragma unroll
        for (int w = 0; w < NTHREADS / 32; ++w) t += smem[w];
        out[0] = fmaxf(t, 0.0f);    // outer ReLU on the scalar
    }
}

extern "C" void kernel_launch(void* const* d_in, const int* in_sizes, int n_in,
                              void* d_out, int out_size, void* d_ws, size_t ws_size,
                              hipStream_t stream) {
    const float* x   = (const float*)d_in[0];
    float*       out = (float*)d_out;
    float*       partials = (float*)d_ws;       // NBLOCKS * 4 bytes = 4 KB

    const long long n  = (long long)in_sizes[0];
    const long long n4 = n >> 2;

    kan_reduce_stage1<<<NBLOCKS, NTHREADS, 0, stream>>>(x, partials, n4);
    kan_reduce_stage2<<<1, NTHREADS, 0, stream>>>(partials, NBLOCKS, x, n, n4, out);
}